// OSS_source_10213432230013
// MI455X (gfx1250) — compile-verified
//
#include <hip/hip_runtime.h>
#include <hip/hip_bf16.h>
#include <math.h>

// ---------------------------------------------------------------------------
// LinOSS-IM forward, fp32, CDNA5 (gfx1250).
//   K1: h = x@W_in^T + b_in ; Bu_{re,im} = h@B_{re,im}^T        (WMMA f32 16x16x4)
//   K2: per-chunk local scan (time-invariant 2x2 M per channel) (VALU)
//   K3: chunk-carry prefix using M^LC (repeated squaring)       (VALU)
//   K4: homogeneous correction  M^k * s_in added to each chunk  (VALU)
//   K5: y = ys_re@Cre^T - ys_im@Cim^T + D*h ; out = y@W_out^T+b (WMMA f32 16x16x4)
// Workspace (floats): h plane (8M) | Bu 2 planes (16M) | ys 2 planes (16M)
//                     | carry_end (128K) | carry_in (128K)  ~= 165 MB total.
// ---------------------------------------------------------------------------

#define BATCH   16
#define LSEQ    4096
#define INDIM   64
#define DMODEL  128
#define OUTDIM  64
#define LC      256            // chunk length (must stay 2^8 for K3 squaring)
#define NC      (LSEQ / LC)    // 16 chunks

typedef float v2f __attribute__((ext_vector_type(2)));
typedef float v8f __attribute__((ext_vector_type(8)));

__device__ __forceinline__ v8f wmma4(v2f a, v2f b, v8f c) {
  // V_WMMA_F32_16X16X4_F32 : D = A(16x4) * B(4x16) + C(16x16), wave32
  return __builtin_amdgcn_wmma_f32_16x16x4_f32(
      /*neg_a=*/false, a, /*neg_b=*/false, b,
      /*c_mod=*/(short)0, c, /*reuse_a=*/false, /*reuse_b=*/false);
}

// Per-channel transition matrix of the implicit LinOSS discretization.
__device__ __forceinline__ void make_M(const float* A_diag, const float* steps,
                                       int p, float& M11, float& M12,
                                       float& M21, float& M22, float& dt) {
  float Aa = fmaxf(A_diag[p], 0.0f);                 // relu
  dt = 1.0f / (1.0f + __expf(-steps[p]));            // sigmoid
  float S = 1.0f / (1.0f + dt * dt * Aa);
  M11 = 1.0f - dt * dt * Aa * S;
  M12 = -dt * Aa * S;
  M21 = dt * S;
  M22 = S;
}

// ---------------------------------------------------------------------------
// K1: input projection + complex B projection, fused through LDS.
// grid = BATCH * (LSEQ/16) blocks, 256 threads (8 waves).
// ---------------------------------------------------------------------------
__global__ __launch_bounds__(256) void oss_k1_inproj(
    const float* __restrict__ x, const float* __restrict__ W_in,
    const float* __restrict__ b_in, const float* __restrict__ Bmat,
    float* __restrict__ hmat, float* __restrict__ bu) {
  __shared__ float lds_h[16][132];                   // padded: stride 132 kills bank conflicts

  int blk   = blockIdx.x;
  int b     = blk >> 8;                              // LSEQ/16 == 256 row tiles
  int row0  = (blk & 255) << 4;
  int lane  = threadIdx.x & 31;
  int wave  = threadIdx.x >> 5;                      // 0..7
  int m     = lane & 15;                             // A-row / B-col / D-col lane index
  int koff  = (lane >> 4) << 1;                      // K offset 0 or 2
  int mbase = (lane >> 4) << 3;                      // D rows: v or v+8

  // ---- stage 1: h tile rows [row0,row0+16), cols [wave*16, wave*16+16)
  const float* xrow = x + ((size_t)b * LSEQ + row0 + m) * INDIM;
  int j = (wave << 4) + m;
  const float* wrow = W_in + (size_t)j * INDIM;
  v8f acc = {};
  for (int k0 = 0; k0 < INDIM; k0 += 4) {
    v2f a, bb;
    a.x  = xrow[k0 + koff];
    a.y  = xrow[k0 + koff + 1];
    bb.x = wrow[k0 + koff];
    bb.y = wrow[k0 + koff + 1];
    acc = wmma4(a, bb, acc);
  }
  float bias = b_in[j];
  float* hout = hmat + ((size_t)b * LSEQ + row0) * DMODEL;
#pragma unroll
  for (int v = 0; v < 8; ++v) {
    int mm = mbase + v;
    float val = acc[v] + bias;
    lds_h[mm][j] = val;
    hout[(size_t)mm * DMODEL + j] = val;
  }
  __syncthreads();

  // ---- stage 2: Bu_{re,im} = h @ B_{re,im}^T ; 16 col-tiles, 2 per wave
  v8f acc2[2] = {{}, {}};
  int ct0 = wave << 1;
  for (int j0 = 0; j0 < DMODEL; j0 += 4) {
    v2f a;
    a.x = lds_h[m][j0 + koff];
    a.y = lds_h[m][j0 + koff + 1];
#pragma unroll
    for (int q = 0; q < 2; ++q) {
      int ct   = ct0 + q;
      int part = ct >> 3;                            // 0 = Re, 1 = Im
      int p    = ((ct & 7) << 4) + m;
      const float* bp = Bmat + ((size_t)p * DMODEL + j0 + koff) * 2 + part;
      v2f bb;
      bb.x = bp[0];
      bb.y = bp[2];
      acc2[q] = wmma4(a, bb, acc2[q]);
    }
  }
#pragma unroll
  for (int q = 0; q < 2; ++q) {
    int ct   = ct0 + q;
    int part = ct >> 3;
    int p    = ((ct & 7) << 4) + m;
    float* bu_out = bu + (((size_t)part * BATCH + b) * LSEQ + row0) * DMODEL + p;
#pragma unroll
    for (int v = 0; v < 8; ++v)
      bu_out[(size_t)(mbase + v) * DMODEL] = acc2[q][v];
  }
}

// ---------------------------------------------------------------------------
// K2: chunk-local scans with zero initial state.
// grid = BATCH*NC blocks, 256 threads (tid -> part, p).
// ---------------------------------------------------------------------------
__global__ __launch_bounds__(256) void oss_k2_scan_local(
    const float* __restrict__ bu, float* __restrict__ ys,
    float* __restrict__ ce, const float* __restrict__ A_diag,
    const float* __restrict__ steps) {
  int blk  = blockIdx.x;
  int b    = blk / NC;
  int c    = blk % NC;
  int part = threadIdx.x >> 7;
  int p    = threadIdx.x & 127;

  float M11, M12, M21, M22, dt;
  make_M(A_diag, steps, p, M11, M12, M21, M22, dt);
  float F1 = M11 * dt, F2 = M21 * dt;

  size_t base = (((size_t)part * BATCH + b) * LSEQ + (size_t)c * LC) * DMODEL + p;
  const float* bp = bu + base;
  float* yp = ys + base;

  float z = 0.0f, y = 0.0f;
  for (int t = 0; t < LC; ++t) {
    float u  = bp[(size_t)t * DMODEL];
    float zn = fmaf(M11, z, fmaf(M12, y, F1 * u));
    float yn = fmaf(M21, z, fmaf(M22, y, F2 * u));
    z = zn;
    y = yn;
    yp[(size_t)t * DMODEL] = y;
  }
  size_t idx  = ((size_t)(b * NC + c) * 2 + part) * DMODEL + p;
  size_t yoff = (size_t)BATCH * NC * 2 * DMODEL;
  ce[idx]        = z;
  ce[yoff + idx] = y;
}

// ---------------------------------------------------------------------------
// K3: chunk-carry prefix. s_in(0)=0 ; s_in(c)=M^LC*s_in(c-1)+end(c-1).
// grid = BATCH blocks, 256 threads.
// ---------------------------------------------------------------------------
__global__ __launch_bounds__(256) void oss_k3_combine(
    const float* __restrict__ ce, float* __restrict__ ci,
    const float* __restrict__ A_diag, const float* __restrict__ steps) {
  int b    = blockIdx.x;
  int part = threadIdx.x >> 7;
  int p    = threadIdx.x & 127;

  float M11, M12, M21, M22, dt;
  make_M(A_diag, steps, p, M11, M12, M21, M22, dt);

  // M^LC with LC = 2^8: eight squarings.
  float P11 = M11, P12 = M12, P21 = M21, P22 = M22;
#pragma unroll
  for (int k = 0; k < 8; ++k) {
    float Q11 = fmaf(P11, P11, P12 * P21);
    float Q12 = fmaf(P11, P12, P12 * P22);
    float Q21 = fmaf(P21, P11, P22 * P21);
    float Q22 = fmaf(P21, P12, P22 * P22);
    P11 = Q11; P12 = Q12; P21 = Q21; P22 = Q22;
  }

  size_t yoff = (size_t)BATCH * NC * 2 * DMODEL;
  float sz = 0.0f, sy = 0.0f;
  for (int c = 0; c < NC; ++c) {
    size_t idx = ((size_t)(b * NC + c) * 2 + part) * DMODEL + p;
    ci[idx]        = sz;
    ci[yoff + idx] = sy;
    float ez = ce[idx], ey = ce[yoff + idx];
    float nz = fmaf(P11, sz, fmaf(P12, sy, ez));
    float ny = fmaf(P21, sz, fmaf(P22, sy, ey));
    sz = nz;
    sy = ny;
  }
}

// ---------------------------------------------------------------------------
// K4: homogeneous correction: ys[t0+k] += (M^{k+1} * s_in).y
// grid = BATCH*NC blocks, 256 threads. Chunk 0 has s_in = 0.
// ---------------------------------------------------------------------------
__global__ __launch_bounds__(256) void oss_k4_correct(
    const float* __restrict__ ci, float* __restrict__ ys,
    const float* __restrict__ A_diag, const float* __restrict__ steps) {
  int blk = blockIdx.x;
  int b   = blk / NC;
  int c   = blk % NC;
  if (c == 0) return;                                // uniform per block
  int part = threadIdx.x >> 7;
  int p    = threadIdx.x & 127;

  float M11, M12, M21, M22, dt;
  make_M(A_diag, steps, p, M11, M12, M21, M22, dt);

  size_t idx  = ((size_t)(b * NC + c) * 2 + part) * DMODEL + p;
  size_t yoff = (size_t)BATCH * NC * 2 * DMODEL;
  float vz = ci[idx];
  float vy = ci[yoff + idx];

  float* yp = ys + (((size_t)part * BATCH + b) * LSEQ + (size_t)c * LC) * DMODEL + p;
  for (int t = 0; t < LC; ++t) {
    float nz = fmaf(M11, vz, M12 * vy);
    float ny = fmaf(M21, vz, M22 * vy);
    vz = nz;
    vy = ny;
    yp[(size_t)t * DMODEL] += vy;
  }
}

// ---------------------------------------------------------------------------
// K5: y = ys_re@Cre^T - ys_im@Cim^T + D*h, then out = y@W_out^T + b_out.
// grid = BATCH * (LSEQ/16) blocks, 256 threads.
// ---------------------------------------------------------------------------
__global__ __launch_bounds__(256) void oss_k5_outproj(
    const float* __restrict__ ys, const float* __restrict__ hmat,
    const float* __restrict__ Cmat, const float* __restrict__ Dvec,
    const float* __restrict__ W_out, const float* __restrict__ b_out,
    float* __restrict__ out) {
  __shared__ float lds_y[16][132];

  int blk   = blockIdx.x;
  int b     = blk >> 8;
  int row0  = (blk & 255) << 4;
  int lane  = threadIdx.x & 31;
  int wave  = threadIdx.x >> 5;
  int m     = lane & 15;
  int koff  = (lane >> 4) << 1;
  int mbase = (lane >> 4) << 3;

  // ---- stage 1: y_model cols [wave*16, wave*16+16)
  int hcol = (wave << 4) + m;
  const float* ys_re = ys + ((size_t)b * LSEQ + row0 + m) * DMODEL;
  const float* ys_im = ys_re + (size_t)BATCH * LSEQ * DMODEL;
  const float* crow  = Cmat + (size_t)hcol * DMODEL * 2;

  v8f acc = {};
  for (int p0 = 0; p0 < DMODEL; p0 += 4) {           // + Re(ys) @ Cre^T
    v2f a, bb;
    a.x  = ys_re[p0 + koff];
    a.y  = ys_re[p0 + koff + 1];
    bb.x = crow[(p0 + koff) * 2];
    bb.y = crow[(p0 + koff + 1) * 2];
    acc = wmma4(a, bb, acc);
  }
  for (int p0 = 0; p0 < DMODEL; p0 += 4) {           // - Im(ys) @ Cim^T
    v2f a, bb;
    a.x  = ys_im[p0 + koff];
    a.y  = ys_im[p0 + koff + 1];
    bb.x = -crow[(p0 + koff) * 2 + 1];
    bb.y = -crow[(p0 + koff + 1) * 2 + 1];
    acc = wmma4(a, bb, acc);
  }
  float Dh = Dvec[hcol];
  const float* hrow = hmat + ((size_t)b * LSEQ + row0) * DMODEL;
#pragma unroll
  for (int v = 0; v < 8; ++v) {
    int mm = mbase + v;
    lds_y[mm][hcol] = fmaf(Dh, hrow[(size_t)mm * DMODEL + hcol], acc[v]);
  }
  __syncthreads();

  // ---- stage 2: out = y_model @ W_out^T + b_out (64 cols -> waves 0..3)
  if (wave < 4) {
    int ocol = (wave << 4) + m;
    const float* wrow = W_out + (size_t)ocol * DMODEL;
    v8f acc2 = {};
    for (int h0 = 0; h0 < DMODEL; h0 += 4) {
      v2f a, bb;
      a.x  = lds_y[m][h0 + koff];
      a.y  = lds_y[m][h0 + koff + 1];
      bb.x = wrow[h0 + koff];
      bb.y = wrow[h0 + koff + 1];
      acc2 = wmma4(a, bb, acc2);
    }
    float bias = b_out[ocol];
    float* orow = out + ((size_t)b * LSEQ + row0) * OUTDIM;
#pragma unroll
    for (int v = 0; v < 8; ++v)
      orow[(size_t)(mbase + v) * OUTDIM + ocol] = acc2[v] + bias;
  }
}

// ---------------------------------------------------------------------------
extern "C" void kernel_launch(void* const* d_in, const int* in_sizes, int n_in,
                              void* d_out, int out_size, void* d_ws, size_t ws_size,
                              hipStream_t stream) {
  (void)in_sizes; (void)n_in; (void)out_size; (void)ws_size;

  const float* x      = (const float*)d_in[0];
  const float* W_in   = (const float*)d_in[1];
  const float* b_in   = (const float*)d_in[2];
  const float* A_diag = (const float*)d_in[3];
  const float* Bmat   = (const float*)d_in[4];
  const float* Cmat   = (const float*)d_in[5];
  const float* Dvec   = (const float*)d_in[6];
  const float* steps  = (const float*)d_in[7];
  const float* W_out  = (const float*)d_in[8];
  const float* b_out  = (const float*)d_in[9];
  float* out = (float*)d_out;
  float* ws  = (float*)d_ws;

  const size_t plane = (size_t)BATCH * LSEQ * DMODEL;   // 8,388,608 floats
  float* hmat = ws;                                     // 1 plane
  float* bu   = ws + plane;                             // 2 planes (Re, Im)
  float* ysb  = ws + plane * 3;                         // 2 planes (Re, Im)
  float* ce   = ws + plane * 5;                         // carry ends  (z|y)
  float* ci   = ce + (size_t)2 * BATCH * NC * 2 * DMODEL; // carry ins (z|y)

  oss_k1_inproj<<<BATCH * (LSEQ / 16), 256, 0, stream>>>(x, W_in, b_in, Bmat,
                                                         hmat, bu);
  oss_k2_scan_local<<<BATCH * NC, 256, 0, stream>>>(bu, ysb, ce, A_diag, steps);
  oss_k3_combine<<<BATCH, 256, 0, stream>>>(ce, ci, A_diag, steps);
  oss_k4_correct<<<BATCH * NC, 256, 0, stream>>>(ci, ysb, A_diag, steps);
  oss_k5_outproj<<<BATCH * (LSEQ / 16), 256, 0, stream>>>(ysb, hmat, Cmat, Dvec,
                                                          W_out, b_out, out);
}